// BNMorph_2044404433290
// MI455X (gfx1250) — compile-verified
//
#include <hip/hip_runtime.h>
#include <math.h>

#define Hh 256
#define Ww 832
#define Bb 2
#define NPIX (Hh * Ww)          // per-image pixels
#define NTOT (Bb * NPIX)        // per-buffer elements
#define NOFF 105                // search offsets
#define RAD 10                  // sense range
#define TS 16                   // WMMA tile side
#define TW 36                   // TS + 2*RAD
#define NCH 9                   // 36/4 K-chunks per row
#define NDY 21                  // row offsets
#define NPAIR (NDY * NCH)       // 189 B-fragments

typedef __attribute__((ext_vector_type(2))) float v2f;
typedef __attribute__((ext_vector_type(8))) float v8f;

// ---------------------------------------------------------------------------
// Setup: banded-Toeplitz weight fragments for V_WMMA_F32_16X16X4_F32,
// stored in exact per-lane layout: lane L holds B[2*(L>>4)+{0,1}][L&15].
// B_{dy,c}[k][n] = w(dy, 4c+k-n-10) if |4c+k-n-10| <= 10 else 0,
// w(dy,dx) = exp(-sqrt(dx^2+dy^2)/5).
// ---------------------------------------------------------------------------
__global__ void k_bfrag(float* __restrict__ bfrag) {
    int idx = blockIdx.x * blockDim.x + threadIdx.x;
    if (idx >= NPAIR * 32) return;
    int lane = idx & 31;
    int p    = idx >> 5;
    int c    = p % NCH;
    int dyl  = p / NCH;
    int dy   = dyl - RAD;
    int n    = lane & 15;
    int kb   = (lane >> 4) * 2;
    float v0 = 0.f, v1 = 0.f;
    {
        int col = 4 * c + kb + 0;
        int dx = col - n - RAD;
        if (dx >= -RAD && dx <= RAD)
            v0 = __expf(-sqrtf((float)(dx * dx + dy * dy)) * 0.2f);
    }
    {
        int col = 4 * c + kb + 1;
        int dx = col - n - RAD;
        if (dx >= -RAD && dx <= RAD)
            v1 = __expf(-sqrtf((float)(dx * dx + dy * dy)) * 0.2f);
    }
    v2f v; v.x = v0; v.y = v1;
    ((v2f*)bfrag)[p * 32 + lane] = v;
}

// ---------------------------------------------------------------------------
// Stage 1: 3x3 box blur with periodic (jnp.roll) boundary.
// ---------------------------------------------------------------------------
__global__ void k_box(const float* __restrict__ in, float* __restrict__ out) {
    int idx = blockIdx.x * blockDim.x + threadIdx.x;
    if (idx >= NTOT) return;
    int j = idx % Ww;
    int i = (idx / Ww) % Hh;
    int b = idx / NPIX;
    const float* m = in + (size_t)b * NPIX;
    float s = 0.f;
    #pragma unroll
    for (int di = -1; di <= 1; ++di) {
        int ii = i + di; ii = (ii < 0) ? ii + Hh : (ii >= Hh ? ii - Hh : ii);
        #pragma unroll
        for (int dj = -1; dj <= 1; ++dj) {
            int jj = j + dj; jj = (jj < 0) ? jj + Ww : (jj >= Ww ? jj - Ww : jj);
            s += m[ii * Ww + jj];
        }
    }
    out[idx] = s;
}

// ---------------------------------------------------------------------------
// Stage 2: orientation = atan2(d/dy, d/dx) of blurred map (jnp.gradient:
// central differences interior, one-sided edges).
// ---------------------------------------------------------------------------
__global__ void k_theta(const float* __restrict__ bmap, float* __restrict__ theta) {
    int idx = blockIdx.x * blockDim.x + threadIdx.x;
    if (idx >= NTOT) return;
    int j = idx % Ww;
    int i = (idx / Ww) % Hh;
    int b = idx / NPIX;
    const float* bb = bmap + (size_t)b * NPIX;
    float gy, gx;
    if (i == 0)            gy = bb[1 * Ww + j] - bb[0 * Ww + j];
    else if (i == Hh - 1)  gy = bb[(Hh - 1) * Ww + j] - bb[(Hh - 2) * Ww + j];
    else                   gy = 0.5f * (bb[(i + 1) * Ww + j] - bb[(i - 1) * Ww + j]);
    if (j == 0)            gx = bb[i * Ww + 1] - bb[i * Ww + 0];
    else if (j == Ww - 1)  gx = bb[i * Ww + Ww - 1] - bb[i * Ww + Ww - 2];
    else                   gx = 0.5f * (bb[i * Ww + j + 1] - bb[i * Ww + j - 1]);
    theta[idx] = atan2f(gy, gx);
}

// ---------------------------------------------------------------------------
// Stage 3: sparsify (keep min-linear-index edge in 5x5 window) + windowed
// correspondence search -> sparse displacement field (dispx, dispy, mask).
// ---------------------------------------------------------------------------
__global__ void k_disp(const float* __restrict__ src, const float* __restrict__ dst,
                       const float* __restrict__ ths, const float* __restrict__ thd,
                       const int* __restrict__ xx, const int* __restrict__ yy,
                       float* __restrict__ dpx, float* __restrict__ dpy,
                       float* __restrict__ pmk) {
    int idx = blockIdx.x * blockDim.x + threadIdx.x;
    if (idx >= NTOT) return;
    int j = idx % Ww;
    int i = (idx / Ww) % Hh;
    int b = idx / NPIX;
    const float* s = src + (size_t)b * NPIX;
    const float* d = dst + (size_t)b * NPIX;
    float ox = 0.f, oy = 0.f, om = 0.f;
    if (s[i * Ww + j] > 0.5f) {
        // keep iff no edge pixel with smaller row-major index in 5x5 window
        bool keep = true;
        for (int di = -2; di <= 2 && keep; ++di) {
            int ii = i + di;
            if (ii < 0 || ii >= Hh) continue;
            for (int dj = -2; dj <= 2; ++dj) {
                int jj = j + dj;
                if (jj < 0 || jj >= Ww) continue;
                if ((ii < i || (ii == i && jj < j)) && s[ii * Ww + jj] > 0.5f) {
                    keep = false; break;
                }
            }
        }
        if (keep) {
            float t0 = ths[(size_t)b * NPIX + i * Ww + j];
            float best = 1e9f;
            int bk = -1;
            for (int k = 0; k < NOFF; ++k) {
                int oyk = yy[k], oxk = xx[k];
                int ii = i + oyk, jj = j + oxk;
                if (ii < 0 || ii >= Hh || jj < 0 || jj >= Ww) continue;
                if (d[ii * Ww + jj] > 0.5f) {
                    float fx = (float)oxk, fy = (float)oyk;
                    float ang = 1.0f - cosf(t0 - thd[(size_t)b * NPIX + ii * Ww + jj]);
                    float sc = 20.0f * sqrtf(fx * fx + fy * fy) + 10.5f * ang;
                    if (sc < best) { best = sc; bk = k; }  // first-min like argmin
                }
            }
            if (bk >= 0 && best < 5e8f) {
                ox = (float)xx[bk]; oy = (float)yy[bk]; om = 1.0f;
            }
        }
    }
    dpx[idx] = ox; dpy[idx] = oy; pmk[idx] = om;
}

// ---------------------------------------------------------------------------
// Stage 4: 21x21 weighted diffusion via WMMA fp32 Toeplitz GEMM.
// One wave per 16x16 output tile; 3 channels accumulated in 3 v8f frags.
// ---------------------------------------------------------------------------
__global__ __launch_bounds__(32) void k_diffuse(
    const float* __restrict__ dpx, const float* __restrict__ dpy,
    const float* __restrict__ pmk, const float* __restrict__ bfrag,
    float* __restrict__ out) {
    __shared__ __align__(16) float tile[3][TW][TW];
    const int lane = threadIdx.x;
    const int j0 = blockIdx.x * TS;
    const int i0 = blockIdx.y * TS;
    const int b  = blockIdx.z;

    auto stage = [&](int ch, const float* __restrict__ src) {
        for (int t = lane; t < TW * TW; t += 32) {
            int r = t / TW, c = t % TW;
            int gi = i0 - RAD + r, gj = j0 - RAD + c;
            float v = 0.f;
            if (gi >= 0 && gi < Hh && gj >= 0 && gj < Ww) v = src[gi * Ww + gj];
            tile[ch][r][c] = v;
        }
    };
    stage(0, dpx + (size_t)b * NPIX);
    stage(1, dpy + (size_t)b * NPIX);
    stage(2, pmk + (size_t)b * NPIX);
    __syncthreads();

    v8f acc0 = {}; v8f acc1 = {}; v8f acc2 = {};
    const int m  = lane & 15;          // A-fragment row
    const int kb = (lane >> 4) * 2;    // A/B-fragment K base
    const v2f* __restrict__ bp = (const v2f*)bfrag;

    for (int dy = 0; dy < NDY; ++dy) {
        #pragma unroll 3
        for (int c = 0; c < NCH; ++c) {
            v2f bf = bp[(dy * NCH + c) * 32 + lane];   // coalesced, L2-hot
            int col = 4 * c + kb;
            v2f a0 = *(const v2f*)&tile[0][m + dy][col];
            v2f a1 = *(const v2f*)&tile[1][m + dy][col];
            v2f a2 = *(const v2f*)&tile[2][m + dy][col];
            acc0 = __builtin_amdgcn_wmma_f32_16x16x4_f32(false, a0, false, bf,
                                                         (short)0, acc0, false, false);
            acc1 = __builtin_amdgcn_wmma_f32_16x16x4_f32(false, a1, false, bf,
                                                         (short)0, acc1, false, false);
            acc2 = __builtin_amdgcn_wmma_f32_16x16x4_f32(false, a2, false, bf,
                                                         (short)0, acc2, false, false);
        }
    }

    // C/D layout: VGPR e = row (e + 8*(lane>>4)), col (lane&15)
    const int n = lane & 15;
    const int mrow = (lane >> 4) * 8;
    float* __restrict__ outx = out + ((size_t)b * 2 + 0) * NPIX;
    float* __restrict__ outy = out + ((size_t)b * 2 + 1) * NPIX;
    #pragma unroll
    for (int e = 0; e < 8; ++e) {
        int i = i0 + mrow + e;
        int j = j0 + n;
        float den = acc2[e];
        float inv = 0.6f / (den + 1e-6f);   // ALPHA_PAD / (den + eps)
        outx[i * Ww + j] = (float)j + acc0[e] * inv;
        outy[i * Ww + j] = (float)i + acc1[e] * inv;
    }
}

// ---------------------------------------------------------------------------
extern "C" void kernel_launch(void* const* d_in, const int* in_sizes, int n_in,
                              void* d_out, int out_size, void* d_ws, size_t ws_size,
                              hipStream_t stream) {
    (void)in_sizes; (void)n_in; (void)out_size; (void)ws_size;
    const float* src = (const float*)d_in[0];
    const float* dst = (const float*)d_in[1];
    const int*   xx  = (const int*)d_in[2];
    const int*   yy  = (const int*)d_in[3];
    float* out = (float*)d_out;

    // workspace layout (floats): th_s | th_d | tmpA(box_s->dispx) |
    //                            tmpB(box_d->dispy) | pm | bfrag
    float* ws   = (float*)d_ws;
    float* th_s = ws;
    float* th_d = ws + (size_t)NTOT;
    float* tmpA = ws + (size_t)2 * NTOT;   // box(src), later dispx
    float* tmpB = ws + (size_t)3 * NTOT;   // box(dst), later dispy
    float* pm   = ws + (size_t)4 * NTOT;
    float* bfr  = ws + (size_t)5 * NTOT;   // NPAIR*32*2 floats (48 KB)

    const int T = 256;
    const int gN = (NTOT + T - 1) / T;

    k_bfrag<<<(NPAIR * 32 + T - 1) / T, T, 0, stream>>>(bfr);
    k_box  <<<gN, T, 0, stream>>>(src, tmpA);
    k_box  <<<gN, T, 0, stream>>>(dst, tmpB);
    k_theta<<<gN, T, 0, stream>>>(tmpA, th_s);
    k_theta<<<gN, T, 0, stream>>>(tmpB, th_d);
    k_disp <<<gN, T, 0, stream>>>(src, dst, th_s, th_d, xx, yy, tmpA, tmpB, pm);
    dim3 grid(Ww / TS, Hh / TS, Bb);   // 52 x 16 x 2 tiles, 1 wave each
    k_diffuse<<<grid, 32, 0, stream>>>(tmpA, tmpB, pm, bfr, out);
}